// MultiheadAttention_515396076376
// MI455X (gfx1250) — compile-verified
//
#include <hip/hip_runtime.h>
#include <hip/hip_bf16.h>

typedef __attribute__((ext_vector_type(2))) float v2f;
typedef __attribute__((ext_vector_type(8))) float v8f;

#define BQ   16
#define QLEN 512
#define KLEN 512
#define DMOD 256
#define NH   8
#define DH   32
#define ALPHA 0.2f

// ---------------------------------------------------------------------------
// Kernel 1a: Vq[h,d] = sum_j Wa[h,j]   * Wq[h*32+j, d]
//            Vk[h,d] = sum_j Wa[h,32+j]* Wk[h*32+j, d]
__global__ void __launch_bounds__(256) prep_v_kernel(
    const float* __restrict__ Wq, const float* __restrict__ Wk,
    const float* __restrict__ Wa, float* __restrict__ Vq, float* __restrict__ Vk) {
  int h = blockIdx.x;
  int d = threadIdx.x;
  float aq = 0.f, ak = 0.f;
  for (int j = 0; j < DH; ++j) {
    aq += Wa[h * (2 * DH) + j]      * Wq[(h * DH + j) * DMOD + d];
    ak += Wa[h * (2 * DH) + DH + j] * Wk[(h * DH + j) * DMOD + d];
  }
  Vq[h * DMOD + d] = aq;
  Vk[h * DMOD + d] = ak;
}

// ---------------------------------------------------------------------------
// Kernel 1b: WoT[k][n] = Wo[n][k]
__global__ void __launch_bounds__(256) transpose_wo_kernel(
    const float* __restrict__ Wo, float* __restrict__ WoT) {
  int n = blockIdx.x, k = threadIdx.x;
  WoT[(size_t)k * DMOD + n] = Wo[(size_t)n * DMOD + k];
}

// ---------------------------------------------------------------------------
// Kernel 2: sq[row,h] = query[row,:]·Vq[h,:]; sk[row,h] = key[row,:]·Vk[h,:]
__global__ void __launch_bounds__(256) sqsk_kernel(
    const float* __restrict__ query, const float* __restrict__ key,
    const float* __restrict__ Vq, const float* __restrict__ Vk,
    float* __restrict__ sqO, float* __restrict__ skO) {
  int idx = blockIdx.x * 256 + threadIdx.x;
  int which = idx >> 16;           // 0 -> sq, 1 -> sk
  int t = idx & 65535;
  int h = t & (NH - 1);
  int row = t >> 3;
  const float* src = which ? key : query;
  const float* V   = which ? Vk  : Vq;
  float acc = 0.f;
  const float* r = src + (size_t)row * DMOD;
  const float* v = V + h * DMOD;
  for (int d = 0; d < DMOD; ++d) acc += r[d] * v[d];
  (which ? skO : sqO)[t] = acc;
}

// ---------------------------------------------------------------------------
// Kernel 3: fused score -> softmax(k) -> attn @ key_head, per (b, q-tile of 16).
// 8 waves/block, wave w == head w. Key blocks are double-buffered in LDS and
// filled with GLOBAL_LOAD_ASYNC_TO_LDS_B128 (ASYNCcnt) overlapping the WMMAs.
#define ADJ_STRIDE 516   // 516 % 64 == 4  -> conflict-free across 16 rows
#define KEY_STRIDE 264   // (2*264) % 64 == 16 -> half-waves hit disjoint banks
#define KB 32            // k rows per staged block
#define NBLK (KLEN / KB) // 16
__global__ void __launch_bounds__(256, 1) attn_agg_kernel(
    const float* __restrict__ key, const int* __restrict__ adj,
    const float* __restrict__ sqG, const float* __restrict__ skG,
    float* __restrict__ agg) {
  int wg = blockIdx.x;          // 16 * 32
  int b  = wg >> 5;
  int q0 = (wg & 31) * 16;
  int tid = threadIdx.x;
  int h    = tid >> 5;          // wave id == head
  int lane = tid & 31;

  __shared__ float skS[KLEN * NH];              // 16 KB
  __shared__ int   adjS[16 * ADJ_STRIDE];       // 33 KB
  __shared__ float keyS[2 * KB * KEY_STRIDE];   // 2 x 33.8 KB

  // async-stage one KB-row key block into LDS buffer `buf`
  auto stage = [&](int blk, int buf) {
    const float* gbase = key + ((size_t)(b * KLEN + blk * KB)) * DMOD;
    float* lbase = &keyS[buf * KB * KEY_STRIDE];
    for (int i = tid; i < KB * 64; i += 256) {  // KB*64 float4 per block
      int r = i >> 6, cc = (i & 63) << 2;
      const float* g = gbase + (size_t)r * DMOD + cc;
      unsigned l = (unsigned)(size_t)(lbase + r * KEY_STRIDE + cc);
      asm volatile("global_load_async_to_lds_b128 %0, %1, off"
                   :: "v"(l), "v"(g) : "memory");
    }
  };

  stage(0, 0);  // block 0 DMA overlaps sk/adj staging and pass 1

  for (int i = tid; i < KLEN * NH; i += 256) skS[i] = skG[(size_t)b * KLEN * NH + i];
  for (int i = tid; i < 16 * KLEN; i += 256) {
    int r = i >> 9, kk = i & 511;
    adjS[r * ADJ_STRIDE + kk] = adj[((size_t)(b * QLEN + q0 + r)) * KLEN + kk];
  }
  __syncthreads();

  int row = lane & 15;
  float sqv = sqG[((size_t)(b * QLEN + q0 + row)) * NH + h];

  // ---- pass 1a: per-row max over k (halves split across lane groups) ----
  int kbase = (lane < 16) ? 0 : 256;
  float m = -3.402823e38f;
  for (int i = 0; i < 256; ++i) {
    int kk = kbase + i;
    float s = sqv + skS[kk * NH + h];
    s = (s >= 0.f) ? s : ALPHA * s;
    s = adjS[row * ADJ_STRIDE + kk] ? s : -1e-8f;
    m = fmaxf(m, s);
  }
  float mc = fmaxf(m, __shfl_xor(m, 16, 32));

  // ---- pass 1b: sum of exp(s - mc); independent accumulation, no rescale ----
  float Z = 0.f;
  for (int i = 0; i < 256; ++i) {
    int kk = kbase + i;
    float s = sqv + skS[kk * NH + h];
    s = (s >= 0.f) ? s : ALPHA * s;
    s = adjS[row * ADJ_STRIDE + kk] ? s : -1e-8f;
    Z += __expf(s - mc);
  }
  float Zc = Z + __shfl_xor(Z, 16, 32);
  float invZ = __builtin_amdgcn_rcpf(Zc);

  // wait for our block-0 async copies, then sync all threads
  asm volatile("s_wait_asynccnt 0x0" ::: "memory");
  __syncthreads();

  // ---- pass 2: exp-weights as WMMA A-tiles, accumulate attn @ key_h ----
  v8f c0 = {}; v8f c1 = {};
  int koff = (lane < 16) ? 0 : 2;      // 16x16x4 f32 layout: VGPR0<->K+0/K+2
  int col  = lane & 15;
  int hcol = h * DH;
  for (int ib = 0; ib < NBLK; ++ib) {
    if (ib + 1 < NBLK) stage(ib + 1, (ib + 1) & 1);   // DMA next block now
    const float* kbuf = &keyS[(ib & 1) * KB * KEY_STRIDE];
    int kg0 = ib * KB;
    for (int kc = 0; kc < KB; kc += 4) {
      int k0 = kc + koff;              // local row in key block
      int kg = kg0 + k0;               // global k
      float s0 = sqv + skS[kg * NH + h];
      s0 = (s0 >= 0.f) ? s0 : ALPHA * s0;
      s0 = adjS[row * ADJ_STRIDE + kg] ? s0 : -1e-8f;
      float s1 = sqv + skS[(kg + 1) * NH + h];
      s1 = (s1 >= 0.f) ? s1 : ALPHA * s1;
      s1 = adjS[row * ADJ_STRIDE + kg + 1] ? s1 : -1e-8f;
      v2f a = { __expf(s0 - mc), __expf(s1 - mc) };
      v2f b0 = { kbuf[k0 * KEY_STRIDE + hcol + col],
                 kbuf[(k0 + 1) * KEY_STRIDE + hcol + col] };
      v2f b1 = { kbuf[k0 * KEY_STRIDE + hcol + 16 + col],
                 kbuf[(k0 + 1) * KEY_STRIDE + hcol + 16 + col] };
      c0 = __builtin_amdgcn_wmma_f32_16x16x4_f32(false, a, false, b0, (short)0, c0, false, false);
      c1 = __builtin_amdgcn_wmma_f32_16x16x4_f32(false, a, false, b1, (short)0, c1, false, false);
    }
    asm volatile("s_wait_asynccnt 0x0" ::: "memory");  // our next-block DMA done
    __syncthreads();                                    // everyone done with cur
  }

  // ---- normalize by 1/Z and store 16x32 head tile ----
  for (int r = 0; r < 8; ++r) {
    int rr = r + ((lane < 16) ? 0 : 8);        // C/D layout: VGPR r -> rows r, r+8
    float iz = __shfl(invZ, rr, 32);
    size_t base = ((size_t)(b * QLEN + q0 + rr)) * DMOD + hcol;
    agg[base + col]      = c0[r] * iz;
    agg[base + 16 + col] = c1[r] * iz;
  }
}

// ---------------------------------------------------------------------------
// Kernel 4: out = query + tanh(agg @ Wo^T). One wave per 16x16 tile, K=256.
#define A_STRIDE 260    // 260 % 64 == 4 -> conflict-free column reads
__global__ void __launch_bounds__(256, 1) outproj_kernel(
    const float* __restrict__ agg, const float* __restrict__ WoT,
    const float* __restrict__ query, float* __restrict__ out) {
  int wg = blockIdx.x;            // 512 m-tiles * 2 halves
  int m0 = (wg >> 1) * 16;
  int half = wg & 1;
  int tid = threadIdx.x;
  int wave = tid >> 5, lane = tid & 31;
  int n0 = (half * 8 + wave) * 16;

  __shared__ float aS[16 * A_STRIDE];
  for (int i = tid; i < 16 * 64; i += 256) {
    int r = i >> 6, cc = (i & 63) << 2;
    const float4 v = *(const float4*)(agg + ((size_t)(m0 + r)) * DMOD + cc);
    *(float4*)(&aS[r * A_STRIDE + cc]) = v;
  }
  __syncthreads();

  int row = lane & 15;
  int koff = (lane < 16) ? 0 : 2;
  int col = lane & 15;
  v8f c = {};
#pragma unroll 4
  for (int kc = 0; kc < DMOD; kc += 4) {
    int k0 = kc + koff;
    v2f a  = { aS[row * A_STRIDE + k0], aS[row * A_STRIDE + k0 + 1] };
    v2f bm = { WoT[(size_t)k0 * DMOD + n0 + col],
               WoT[(size_t)(k0 + 1) * DMOD + n0 + col] };
    c = __builtin_amdgcn_wmma_f32_16x16x4_f32(false, a, false, bm, (short)0, c, false, false);
  }
  for (int r = 0; r < 8; ++r) {
    int rr = r + ((lane < 16) ? 0 : 8);
    size_t idx = ((size_t)(m0 + rr)) * DMOD + n0 + col;
    out[idx] = query[idx] + tanhf(c[r]);
  }
}

// ---------------------------------------------------------------------------
extern "C" void kernel_launch(void* const* d_in, const int* in_sizes, int n_in,
                              void* d_out, int out_size, void* d_ws, size_t ws_size,
                              hipStream_t stream) {
  const float* query = (const float*)d_in[0];
  const float* key   = (const float*)d_in[1];
  const int*   adj   = (const int*)d_in[2];
  const float* Wq    = (const float*)d_in[3];
  const float* Wk    = (const float*)d_in[4];
  const float* Wa    = (const float*)d_in[5];
  const float* Wo    = (const float*)d_in[6];
  float* out = (float*)d_out;

  float* ws  = (float*)d_ws;
  float* Vq  = ws;                 // 2048
  float* Vk  = ws + 2048;          // 2048
  float* WoT = ws + 4096;          // 65536
  float* sqB = ws + 69632;         // 65536
  float* skB = ws + 135168;        // 65536
  float* agg = ws + 200704;        // 2097152  (total ~9.2 MB)

  prep_v_kernel<<<NH, 256, 0, stream>>>(Wq, Wk, Wa, Vq, Vk);
  transpose_wo_kernel<<<DMOD, 256, 0, stream>>>(Wo, WoT);
  sqsk_kernel<<<512, 256, 0, stream>>>(query, key, Vq, Vk, sqB, skB);
  attn_agg_kernel<<<BQ * (QLEN / 16), 256, 0, stream>>>(key, adj, sqB, skB, agg);
  outproj_kernel<<<(BQ * QLEN / 16) * 2, 256, 0, stream>>>(agg, WoT, query, out);
}